// ConvBertMixedAttention_24558622999249
// MI455X (gfx1250) — compile-verified
//
#include <hip/hip_runtime.h>
#include <hip/hip_bf16.h>
#include <math.h>

typedef __attribute__((ext_vector_type(2))) float v2f;
typedef __attribute__((ext_vector_type(8))) float v8f;

namespace cfg {
constexpr int B    = 4;
constexpr int S    = 2048;
constexpr int HID  = 768;
constexpr int H    = 6;
constexpr int D    = 64;
constexpr int AHS  = 384;   // H*D
constexpr int KT   = 9;     // conv taps
constexpr int HK   = 54;    // H*KT
constexpr int HALF = 384;   // HID/2
constexpr int BS   = B * S; // 8192 token rows
}

__device__ __forceinline__ v8f wmma4(v2f a, v2f b, v8f c) {
  // (neg_a, A, neg_b, B, c_mod, C, reuse_a, reuse_b)
  return __builtin_amdgcn_wmma_f32_16x16x4_f32(false, a, false, b, (short)0, c,
                                               false, false);
}

// Load the four B fragments (4x16 each) for one k-step.
//   TRANSB: W is [N, Kd] row-major (b pair contiguous -> b64 loads)
//  !TRANSB: W is [Kd, N] row-major
template <bool TRANSB>
__device__ __forceinline__ void load_b4(const float* __restrict__ W, int ldw,
                                        int n0lt, int koff, int kk,
                                        v2f& b0, v2f& b1, v2f& b2, v2f& b3) {
  if (TRANSB) {
    const float* w = W + (size_t)n0lt * ldw + kk + koff;
    b0.x = w[0]; b0.y = w[1]; w += (size_t)16 * ldw;
    b1.x = w[0]; b1.y = w[1]; w += (size_t)16 * ldw;
    b2.x = w[0]; b2.y = w[1]; w += (size_t)16 * ldw;
    b3.x = w[0]; b3.y = w[1];
  } else {
    const float* w0 = W + (size_t)(kk + koff) * ldw + n0lt;
    const float* w1 = w0 + ldw;
    b0.x = w0[0];  b0.y = w1[0];
    b1.x = w0[16]; b1.y = w1[16];
    b2.x = w0[32]; b2.y = w1[32];
    b3.x = w0[48]; b3.y = w1[48];
  }
}

#define WMMA8()                                                                \
  do {                                                                         \
    acc0 = wmma4(a0, b0, acc0);                                                \
    acc1 = wmma4(a0, b1, acc1);                                                \
    acc2 = wmma4(a0, b2, acc2);                                                \
    acc3 = wmma4(a0, b3, acc3);                                                \
    acc4 = wmma4(a1, b0, acc4);                                                \
    acc5 = wmma4(a1, b1, acc5);                                                \
    acc6 = wmma4(a1, b2, acc6);                                                \
    acc7 = wmma4(a1, b3, acc7);                                                \
  } while (0)

// ---------------------------------------------------------------------------
// Generic fp32 WMMA GEMM: C[m,n] = alpha * sum_k A[m,k]*W(k,n) + bias[n]
// Block = 128 threads (4 waves). Wave tile = 32 rows x 64 cols (8 acc).
// Software-pipelined: k+4 operands prefetched while k's WMMAs issue.
// Requires: M % 128 == 0, N % 64 == 0, Kd % 4 == 0, Kd >= 8.
// ---------------------------------------------------------------------------
template <bool TRANSB>
__global__ __launch_bounds__(128) void gemm_wmma_k(
    const float* __restrict__ A, int lda,
    const float* __restrict__ W, int ldw,
    const float* __restrict__ bias, float alpha,
    float* __restrict__ C, int ldc, int Kd) {
  const int lane = threadIdx.x & 31;
  const int wave = threadIdx.x >> 5;
  const int lt   = lane & 15;
  const int koff = (lane >> 4) * 2;  // lanes 0-15 -> K{0,1}; 16-31 -> K{2,3}
  const int m0   = blockIdx.y * 128 + wave * 32;
  const int n0   = blockIdx.x * 64;
  const int n0lt = n0 + lt;

  const float* Arow0 = A + (size_t)(m0 + lt) * lda + koff;
  const float* Arow1 = Arow0 + (size_t)16 * lda;

  v8f acc0 = {}, acc1 = {}, acc2 = {}, acc3 = {};
  v8f acc4 = {}, acc5 = {}, acc6 = {}, acc7 = {};

  v2f a0, a1, b0, b1, b2, b3;
  a0.x = Arow0[0]; a0.y = Arow0[1];
  a1.x = Arow1[0]; a1.y = Arow1[1];
  load_b4<TRANSB>(W, ldw, n0lt, koff, 0, b0, b1, b2, b3);

  int kk = 0;
  for (; kk + 4 < Kd; kk += 4) {
    v2f na0, na1, nb0, nb1, nb2, nb3;
    na0.x = Arow0[kk + 4]; na0.y = Arow0[kk + 5];
    na1.x = Arow1[kk + 4]; na1.y = Arow1[kk + 5];
    load_b4<TRANSB>(W, ldw, n0lt, koff, kk + 4, nb0, nb1, nb2, nb3);
    WMMA8();
    a0 = na0; a1 = na1;
    b0 = nb0; b1 = nb1; b2 = nb2; b3 = nb3;
  }
  WMMA8();  // peeled final step

  float bv0 = 0.f, bv1 = 0.f, bv2 = 0.f, bv3 = 0.f;
  if (bias) {
    bv0 = bias[n0lt];
    bv1 = bias[n0lt + 16];
    bv2 = bias[n0lt + 32];
    bv3 = bias[n0lt + 48];
  }
  const int rb0 = m0 + (lane >> 4) * 8;  // C/D: vgpr r -> row r (+8 hi half)
#pragma unroll
  for (int r = 0; r < 8; ++r) {
    float* c0 = C + (size_t)(rb0 + r) * ldc + n0lt;
    c0[0]  = alpha * acc0[r] + bv0;
    c0[16] = alpha * acc1[r] + bv1;
    c0[32] = alpha * acc2[r] + bv2;
    c0[48] = alpha * acc3[r] + bv3;
    float* c1 = c0 + (size_t)16 * ldc;
    c1[0]  = alpha * acc4[r] + bv0;
    c1[16] = alpha * acc5[r] + bv1;
    c1[32] = alpha * acc6[r] + bv2;
    c1[48] = alpha * acc7[r] + bv3;
  }
}

// ---------------------------------------------------------------------------
// Attention scores: scores[z=b*H+h][q][k] = (Qh . Kh)/sqrt(D) + mask[b][k]
// Same pipelined 32x64 wave tile; Kd = D = 64.
// ---------------------------------------------------------------------------
__global__ __launch_bounds__(128) void attn_scores_k(
    const float* __restrict__ qb, const float* __restrict__ kb,
    const float* __restrict__ mask, float* __restrict__ scores) {
  using namespace cfg;
  const int z = blockIdx.z, b = z / H, h = z % H;
  const int lane = threadIdx.x & 31;
  const int wave = threadIdx.x >> 5;
  const int lt = lane & 15;
  const int koff = (lane >> 4) * 2;
  const int m0 = blockIdx.y * 128 + wave * 32;
  const int n0 = blockIdx.x * 64;
  const int n0lt = n0 + lt;

  const float* Arow0 = qb + ((size_t)b * S + m0 + lt) * AHS + h * D + koff;
  const float* Arow1 = Arow0 + (size_t)16 * AHS;
  const float* Wb = kb + (size_t)b * S * AHS + h * D;  // [S][D], ldw = AHS

  v8f acc0 = {}, acc1 = {}, acc2 = {}, acc3 = {};
  v8f acc4 = {}, acc5 = {}, acc6 = {}, acc7 = {};

  v2f a0, a1, b0, b1, b2, b3;
  a0.x = Arow0[0]; a0.y = Arow0[1];
  a1.x = Arow1[0]; a1.y = Arow1[1];
  load_b4<true>(Wb, AHS, n0lt, koff, 0, b0, b1, b2, b3);

  int kk = 0;
  for (; kk + 4 < D; kk += 4) {
    v2f na0, na1, nb0, nb1, nb2, nb3;
    na0.x = Arow0[kk + 4]; na0.y = Arow0[kk + 5];
    na1.x = Arow1[kk + 4]; na1.y = Arow1[kk + 5];
    load_b4<true>(Wb, AHS, n0lt, koff, kk + 4, nb0, nb1, nb2, nb3);
    WMMA8();
    a0 = na0; a1 = na1;
    b0 = nb0; b1 = nb1; b2 = nb2; b3 = nb3;
  }
  WMMA8();

  const float* mrow = mask + (size_t)b * S;
  const float mk0 = mrow[n0lt];
  const float mk1 = mrow[n0lt + 16];
  const float mk2 = mrow[n0lt + 32];
  const float mk3 = mrow[n0lt + 48];
  const float alpha = 0.125f;  // 1/sqrt(64)
  const int rb0 = m0 + (lane >> 4) * 8;
  float* Cz = scores + (size_t)z * S * S;
#pragma unroll
  for (int r = 0; r < 8; ++r) {
    float* c0 = Cz + (size_t)(rb0 + r) * S + n0lt;
    c0[0]  = alpha * acc0[r] + mk0;
    c0[16] = alpha * acc1[r] + mk1;
    c0[32] = alpha * acc2[r] + mk2;
    c0[48] = alpha * acc3[r] + mk3;
    float* c1 = c0 + (size_t)16 * S;
    c1[0]  = alpha * acc4[r] + mk0;
    c1[16] = alpha * acc5[r] + mk1;
    c1[32] = alpha * acc6[r] + mk2;
    c1[48] = alpha * acc7[r] + mk3;
  }
}

// ---------------------------------------------------------------------------
// Row-wise softmax stats over stored scores: per row store (max, 1/sum).
// ---------------------------------------------------------------------------
__global__ __launch_bounds__(256) void softmax_stats_k(
    const float* __restrict__ scores, float* __restrict__ stats) {
  using namespace cfg;
  const size_t row = blockIdx.x;
  const float* r = scores + row * (size_t)S;
  __shared__ float red[256];
  float mx = -INFINITY;
  for (int i = threadIdx.x; i < S; i += 256) mx = fmaxf(mx, r[i]);
  red[threadIdx.x] = mx;
  __syncthreads();
  for (int s = 128; s > 0; s >>= 1) {
    if ((int)threadIdx.x < s)
      red[threadIdx.x] = fmaxf(red[threadIdx.x], red[threadIdx.x + s]);
    __syncthreads();
  }
  mx = red[0];
  __syncthreads();
  float sum = 0.f;
  for (int i = threadIdx.x; i < S; i += 256) sum += __expf(r[i] - mx);
  red[threadIdx.x] = sum;
  __syncthreads();
  for (int s = 128; s > 0; s >>= 1) {
    if ((int)threadIdx.x < s) red[threadIdx.x] += red[threadIdx.x + s];
    __syncthreads();
  }
  if (threadIdx.x == 0) {
    stats[2 * row]     = mx;
    stats[2 * row + 1] = 1.0f / red[0];
  }
}

// ---------------------------------------------------------------------------
// Context: attn_out[b,q,h*D+n] = sum_k softmax(scores)[z,q,k] * v[b,k,h*D+n]
// Probabilities produced on the fly (v_exp trans ops co-execute with WMMA).
// Wave tile 32 rows x 64 cols (all of N); pipelined raw-score prefetch.
// ---------------------------------------------------------------------------
__global__ __launch_bounds__(128) void attn_ctx_k(
    const float* __restrict__ scores, const float* __restrict__ stats,
    const float* __restrict__ vb, float* __restrict__ attn_out) {
  using namespace cfg;
  const int z = blockIdx.z, b = z / H, h = z % H;
  const int lane = threadIdx.x & 31;
  const int wave = threadIdx.x >> 5;
  const int lt = lane & 15;
  const int koff = (lane >> 4) * 2;
  const int m0 = blockIdx.y * 128 + wave * 32;

  const size_t row0 = (size_t)z * S + m0 + lt;
  const size_t row1 = row0 + 16;
  const float* prow0 = scores + row0 * S + koff;
  const float* prow1 = scores + row1 * S + koff;
  const float mx0 = stats[2 * row0], inv0 = stats[2 * row0 + 1];
  const float mx1 = stats[2 * row1], inv1 = stats[2 * row1 + 1];
  const float* Vh = vb + (size_t)b * S * AHS + h * D;

  v8f acc0 = {}, acc1 = {}, acc2 = {}, acc3 = {};
  v8f acc4 = {}, acc5 = {}, acc6 = {}, acc7 = {};

  v2f r0, r1, b0, b1, b2, b3;
  r0.x = prow0[0]; r0.y = prow0[1];
  r1.x = prow1[0]; r1.y = prow1[1];
  load_b4<true>(Vh, AHS, lt, koff, 0, b0, b1, b2, b3);

  int kk = 0;
  for (; kk + 4 < S; kk += 4) {
    v2f nr0, nr1, nb0, nb1, nb2, nb3;
    nr0.x = prow0[kk + 4]; nr0.y = prow0[kk + 5];
    nr1.x = prow1[kk + 4]; nr1.y = prow1[kk + 5];
    load_b4<true>(Vh, AHS, lt, koff, kk + 4, nb0, nb1, nb2, nb3);
    v2f a0, a1;
    a0.x = __expf(r0.x - mx0) * inv0;
    a0.y = __expf(r0.y - mx0) * inv0;
    a1.x = __expf(r1.x - mx1) * inv1;
    a1.y = __expf(r1.y - mx1) * inv1;
    WMMA8();
    r0 = nr0; r1 = nr1;
    b0 = nb0; b1 = nb1; b2 = nb2; b3 = nb3;
  }
  {
    v2f a0, a1;
    a0.x = __expf(r0.x - mx0) * inv0;
    a0.y = __expf(r0.y - mx0) * inv0;
    a1.x = __expf(r1.x - mx1) * inv1;
    a1.y = __expf(r1.y - mx1) * inv1;
    WMMA8();
  }

  const int rb0 = m0 + (lane >> 4) * 8;
#pragma unroll
  for (int r = 0; r < 8; ++r) {
    float* c0 = attn_out + ((size_t)b * S + rb0 + r) * AHS + h * D + lt;
    c0[0]  = acc0[r];
    c0[16] = acc1[r];
    c0[32] = acc2[r];
    c0[48] = acc3[r];
    float* c1 = c0 + (size_t)16 * AHS;
    c1[0]  = acc4[r];
    c1[16] = acc5[r];
    c1[32] = acc6[r];
    c1[48] = acc7[r];
  }
}

// ---------------------------------------------------------------------------
// Depthwise conv over sequence: dw[b,s,c] = sum_t x[b,s+t-4,c] * k[c,t]
// ---------------------------------------------------------------------------
__global__ __launch_bounds__(256) void dwconv_k(
    const float* __restrict__ x, const float* __restrict__ dwk,
    float* __restrict__ out) {
  using namespace cfg;
  const size_t i = (size_t)blockIdx.x * 256 + threadIdx.x;
  if (i >= (size_t)BS * HID) return;
  const int c = (int)(i % HID);
  const size_t bs = i / HID;
  const int s = (int)(bs % S);
  const size_t bbase = (bs / S) * (size_t)S;
  float acc = 0.f;
#pragma unroll
  for (int t = 0; t < KT; ++t) {
    const int sp = s + t - KT / 2;
    if (sp >= 0 && sp < S) acc += x[(bbase + sp) * HID + c] * dwk[c * KT + t];
  }
  out[i] = acc;
}

// ---------------------------------------------------------------------------
// conv filter: logits = (ks_out*q) @ Wak + bak -> softmax over the 9 taps.
// ---------------------------------------------------------------------------
__global__ __launch_bounds__(64) void conv_filter_k(
    const float* __restrict__ ks, const float* __restrict__ qb,
    const float* __restrict__ Wak, const float* __restrict__ bak,
    float* __restrict__ filt) {
  using namespace cfg;
  const size_t bs = blockIdx.x;
  __shared__ float row[AHS];
  __shared__ float logit[HK];
  const float* kr = ks + bs * AHS;
  const float* qr = qb + bs * AHS;
  for (int i = threadIdx.x; i < AHS; i += 64) row[i] = kr[i] * qr[i];
  __syncthreads();
  if ((int)threadIdx.x < HK) {
    float acc = bak[threadIdx.x];
    for (int i = 0; i < AHS; ++i) acc += row[i] * Wak[i * HK + threadIdx.x];
    logit[threadIdx.x] = acc;
  }
  __syncthreads();
  if ((int)threadIdx.x < H) {
    const int h = threadIdx.x;
    float mx = -INFINITY;
#pragma unroll
    for (int t = 0; t < KT; ++t) mx = fmaxf(mx, logit[h * KT + t]);
    float e[KT];
    float sm = 0.f;
#pragma unroll
    for (int t = 0; t < KT; ++t) {
      e[t] = __expf(logit[h * KT + t] - mx);
      sm += e[t];
    }
    const float inv = 1.0f / sm;
#pragma unroll
    for (int t = 0; t < KT; ++t) filt[bs * HK + h * KT + t] = e[t] * inv;
  }
}

// ---------------------------------------------------------------------------
// conv_out[b,s,o] = sum_t v[b, s+t-4, o] * filt[b,s, (o/64)*9 + t]
// ---------------------------------------------------------------------------
__global__ __launch_bounds__(256) void conv_out_k(
    const float* __restrict__ vb, const float* __restrict__ filt,
    float* __restrict__ out) {
  using namespace cfg;
  const size_t i = (size_t)blockIdx.x * 256 + threadIdx.x;
  if (i >= (size_t)BS * AHS) return;
  const int o = (int)(i % AHS);
  const size_t bs = i / AHS;
  const int s = (int)(bs % S);
  const size_t bbase = (bs / S) * (size_t)S;
  const float* f = filt + bs * HK + (o / D) * KT;
  float acc = 0.f;
#pragma unroll
  for (int t = 0; t < KT; ++t) {
    const int sp = s + t - KT / 2;
    if (sp >= 0 && sp < S) acc += vb[(bbase + sp) * AHS + o] * f[t];
  }
  out[i] = acc;
}

// ---------------------------------------------------------------------------
extern "C" void kernel_launch(void* const* d_in, const int* in_sizes, int n_in,
                              void* d_out, int out_size, void* d_ws,
                              size_t ws_size, hipStream_t stream) {
  using namespace cfg;
  (void)in_sizes; (void)n_in; (void)out_size; (void)ws_size;

  const float* X    = (const float*)d_in[0];   // [B,S,HID]
  const float* mask = (const float*)d_in[1];   // [B,S]
  const float* Wq   = (const float*)d_in[2];
  const float* bq   = (const float*)d_in[3];
  const float* Wk   = (const float*)d_in[4];
  const float* bk   = (const float*)d_in[5];
  const float* Wv   = (const float*)d_in[6];
  const float* bv   = (const float*)d_in[7];
  const float* dwk  = (const float*)d_in[8];   // [HID,1,KT]
  const float* pwk  = (const float*)d_in[9];   // [AHS,HID] (out,in)
  const float* sepb = (const float*)d_in[10];
  const float* Wak  = (const float*)d_in[11];  // [AHS,HK]
  const float* bak  = (const float*)d_in[12];
  const float* Wsl  = (const float*)d_in[13];  // [AHS,HALF]
  const float* bsl  = (const float*)d_in[14];
  const float* Wcl  = (const float*)d_in[15];
  const float* bcl  = (const float*)d_in[16];

  float* out      = (float*)d_out;
  float* ctx_out  = out;                                   // [B,S,HID]
  float* attn_out = out + (size_t)BS * HID;                // [B,S,AHS]
  float* scores   = attn_out + (size_t)BS * AHS;           // [B,H,S,S]

  float* ws = (float*)d_ws;
  float* qb    = ws;                        // BS*AHS
  float* kb    = qb + (size_t)BS * AHS;     // BS*AHS
  float* vb    = kb + (size_t)BS * AHS;     // BS*AHS
  float* dwb   = vb + (size_t)BS * AHS;     // BS*HID
  float* ksb   = dwb + (size_t)BS * HID;    // BS*AHS
  float* filtb = ksb + (size_t)BS * AHS;    // BS*HK
  float* convb = filtb + (size_t)BS * HK;   // BS*AHS
  float* statb = convb + (size_t)BS * AHS;  // B*H*S*2

  const dim3 blk(128);
  const dim3 gProj(AHS / 64, BS / 128);  // 6 x 64

  // Q, K, V projections
  gemm_wmma_k<false><<<gProj, blk, 0, stream>>>(X, HID, Wq, AHS, bq, 1.0f, qb, AHS, HID);
  gemm_wmma_k<false><<<gProj, blk, 0, stream>>>(X, HID, Wk, AHS, bk, 1.0f, kb, AHS, HID);
  gemm_wmma_k<false><<<gProj, blk, 0, stream>>>(X, HID, Wv, AHS, bv, 1.0f, vb, AHS, HID);

  // attention scores (written to d_out; third reference output)
  attn_scores_k<<<dim3(S / 64, S / 128, B * H), blk, 0, stream>>>(qb, kb, mask, scores);

  // per-row softmax stats, then P @ V with on-the-fly normalization
  softmax_stats_k<<<dim3(B * H * S), 256, 0, stream>>>(scores, statb);
  attn_ctx_k<<<dim3(1, S / 128, B * H), blk, 0, stream>>>(scores, statb, vb, attn_out);

  // sdconv path
  dwconv_k<<<(int)(((size_t)BS * HID + 255) / 256), 256, 0, stream>>>(X, dwk, dwb);
  gemm_wmma_k<true><<<gProj, blk, 0, stream>>>(dwb, HID, pwk, HID, sepb, 1.0f, ksb, AHS, HID);
  conv_filter_k<<<dim3(BS), 64, 0, stream>>>(ksb, qb, Wak, bak, filtb);
  conv_out_k<<<(int)(((size_t)BS * AHS + 255) / 256), 256, 0, stream>>>(vb, filtb, convb);

  // mix: context = [attn_out @ Wsl + bsl , conv_out @ Wcl + bcl]
  gemm_wmma_k<false><<<dim3(HALF / 64, BS / 128), blk, 0, stream>>>(
      attn_out, AHS, Wsl, HALF, bsl, 1.0f, ctx_out, HID, AHS);
  gemm_wmma_k<false><<<dim3(HALF / 64, BS / 128), blk, 0, stream>>>(
      convb, AHS, Wcl, HALF, bcl, 1.0f, ctx_out + HALF, HID, AHS);
}